// Net_76690936037575
// MI455X (gfx1250) — compile-verified
//
#include <hip/hip_runtime.h>
#include <hip/hip_bf16.h>
#include <math.h>

typedef __attribute__((ext_vector_type(2))) float v2f;
typedef __attribute__((ext_vector_type(8))) float v8f;

#define DIM   128
#define LSTR  132   // padded LDS row stride (floats): bank-conflict-free A fetches
#define WPB   4     // waves per block (block = 128 threads); LDS = 4*16*132*4 = 33.8 KB
#define WMAT  (DIM * DIM)   // 16384 floats per packed matrix

// ---------------- owner scatter (con pass runs after var pass -> con wins) ----
__global__ __launch_bounds__(256) void owner_scatter_var(
    const long long* __restrict__ assoc, int* __restrict__ owner, int n) {
  int i = blockIdx.x * blockDim.x + threadIdx.x;
  if (i < n) owner[assoc[i]] = i;
}
__global__ __launch_bounds__(256) void owner_scatter_con(
    const long long* __restrict__ assoc, int* __restrict__ owner, int n) {
  int i = blockIdx.x * blockDim.x + threadIdx.x;
  if (i < n) owner[assoc[i]] = i | 0x40000000;
}

// ---------------- pack a 128x128 row-major W into WMMA B-operand lane order ---
// P[(((nt*16)+ktp)*32 + lane)*4 + q], q = 2*j + v  ->  W[k*128 + col] with
//   k = ktp*8 + j*4 + 2*(lane>>4) + v,  col = nt*16 + (lane&15)
// so one b128 per lane supplies the B operands of k-tiles 2*ktp and 2*ktp+1.
__global__ __launch_bounds__(256) void pack_w(const float* __restrict__ W,
                                              float* __restrict__ P) {
  int idx  = blockIdx.x * blockDim.x + threadIdx.x;   // 0..16383
  int q    = idx & 3;
  int lane = (idx >> 2) & 31;
  int ktp  = (idx >> 7) & 15;
  int nt   = idx >> 11;
  int j = q >> 1, v = q & 1;
  int k   = ktp * 8 + j * 4 + 2 * (lane >> 4) + v;
  int col = nt * 16 + (lane & 15);
  P[idx] = W[k * DIM + col];
}

// ---------------- one 16x128 @ 128x128 layer via V_WMMA_F32_16X16X4_F32 -------
// Wp is the PACKED weight matrix. src/dst may alias: all K-loop LDS reads
// complete (in-order DS) before the epilogue stores the tile from registers.
__device__ __forceinline__ void layer_wmma(const float* src, float* dst,
                                           const float* __restrict__ Wp,
                                           const float* __restrict__ bias,
                                           int n, int half, int lane, bool doRelu) {
  const float4* W4p = (const float4*)Wp;
  v8f acc[8];
#pragma unroll
  for (int nt = 0; nt < 8; ++nt) {
    float bv = bias[nt * 16 + n];
    v8f t = {bv, bv, bv, bv, bv, bv, bv, bv};
    acc[nt] = t;
  }
#pragma unroll 1
  for (int ktp = 0; ktp < 16; ++ktp) {       // two k-tiles per iteration
    int k0 = ktp * 8 + half * 2;
    v2f a0 = {src[n * LSTR + k0],     src[n * LSTR + k0 + 1]};
    v2f a1 = {src[n * LSTR + k0 + 4], src[n * LSTR + k0 + 5]};
    float4 bq[8];
#pragma unroll
    for (int nt = 0; nt < 8; ++nt)
      bq[nt] = W4p[(nt * 16 + ktp) * 32 + lane];
#pragma unroll
    for (int nt = 0; nt < 8; ++nt) {
      v2f b0 = {bq[nt].x, bq[nt].y};
      acc[nt] = __builtin_amdgcn_wmma_f32_16x16x4_f32(
          false, a0, false, b0, (short)0, acc[nt], false, false);
    }
#pragma unroll
    for (int nt = 0; nt < 8; ++nt) {
      v2f b1 = {bq[nt].z, bq[nt].w};
      acc[nt] = __builtin_amdgcn_wmma_f32_16x16x4_f32(
          false, a1, false, b1, (short)0, acc[nt], false, false);
    }
  }
#pragma unroll
  for (int nt = 0; nt < 8; ++nt) {
#pragma unroll
    for (int v = 0; v < 8; ++v) {
      float d = acc[nt][v];
      if (doRelu) d = fmaxf(d, 0.0f);
      dst[(v + half * 8) * LSTR + nt * 16 + n] = d;   // D: M = v + 8*half, N = col
    }
  }
}

// Mixed var/con tile: two masked passes, per-row bias in epilogue (no relu).
__device__ __forceinline__ void layer_wmma_mixed(const float* src, float* dst,
                                                 const float* __restrict__ Wvp,
                                                 const float* __restrict__ Bv,
                                                 const float* __restrict__ Wcp,
                                                 const float* __restrict__ Bc,
                                                 int n, int half, int lane,
                                                 unsigned conMask) {
  bool myCon = (conMask >> n) & 1u;
  v8f acc[8];
#pragma unroll
  for (int nt = 0; nt < 8; ++nt) { v8f z = {0,0,0,0,0,0,0,0}; acc[nt] = z; }
#pragma unroll 1
  for (int pass = 0; pass < 2; ++pass) {
    const float4* W4p = (const float4*)(pass ? Wcp : Wvp);
    bool zero = pass ? !myCon : myCon;       // zero A rows of the other type
#pragma unroll 1
    for (int ktp = 0; ktp < 16; ++ktp) {
      int k0 = ktp * 8 + half * 2;
      float s0 = src[n * LSTR + k0],     s1 = src[n * LSTR + k0 + 1];
      float s2 = src[n * LSTR + k0 + 4], s3 = src[n * LSTR + k0 + 5];
      v2f a0 = {zero ? 0.0f : s0, zero ? 0.0f : s1};
      v2f a1 = {zero ? 0.0f : s2, zero ? 0.0f : s3};
      float4 bq[8];
#pragma unroll
      for (int nt = 0; nt < 8; ++nt)
        bq[nt] = W4p[(nt * 16 + ktp) * 32 + lane];
#pragma unroll
      for (int nt = 0; nt < 8; ++nt) {
        v2f b0 = {bq[nt].x, bq[nt].y};
        acc[nt] = __builtin_amdgcn_wmma_f32_16x16x4_f32(
            false, a0, false, b0, (short)0, acc[nt], false, false);
      }
#pragma unroll
      for (int nt = 0; nt < 8; ++nt) {
        v2f b1 = {bq[nt].z, bq[nt].w};
        acc[nt] = __builtin_amdgcn_wmma_f32_16x16x4_f32(
            false, a1, false, b1, (short)0, acc[nt], false, false);
      }
    }
  }
#pragma unroll
  for (int nt = 0; nt < 8; ++nt) {
#pragma unroll
    for (int v = 0; v < 8; ++v) {
      int m = v + half * 8;
      bool rc = (conMask >> m) & 1u;
      float bias = (rc ? Bc : Bv)[nt * 16 + n];
      dst[m * LSTR + nt * 16 + n] = acc[nt][v] + bias;
    }
  }
}

// ---------------- fused: gather -> mlp2 -> 3x(128x128+relu) -> sigmoid head --
__global__ __launch_bounds__(WPB * 32, 4) void fused_net(
    const float* __restrict__ varF, const float* __restrict__ conF,
    const long long* __restrict__ assoc_var, const int* __restrict__ owner,
    const float* __restrict__ vW1, const float* __restrict__ vb1,
    const float* __restrict__ Pv2, const float* __restrict__ vb2,
    const float* __restrict__ cW1, const float* __restrict__ cb1,
    const float* __restrict__ Pc2, const float* __restrict__ cb2,
    const float* __restrict__ P1, const float* __restrict__ b1,
    const float* __restrict__ P2, const float* __restrict__ b2,
    const float* __restrict__ P3, const float* __restrict__ b3,
    const float* __restrict__ W4, const float* __restrict__ b4,
    float* __restrict__ out, int n_var, int n_tiles) {
  __shared__ float lds[WPB][16 * LSTR];
  const int wave = threadIdx.x >> 5;
  const int lane = threadIdx.x & 31;
  const int n = lane & 15;       // column / row-id within tile
  const int half = lane >> 4;    // 0 or 1
  float* buf = &lds[wave][0];

  const int waveGlobal = blockIdx.x * WPB + wave;
  const int waveStride = gridDim.x * WPB;

#pragma unroll 1
  for (int tile = waveGlobal; tile < n_tiles; tile += waveStride) {
    // ---- per-row meta: each lane handles row r = n (both halves duplicate) --
    const int i = tile * 16 + n;
    const bool valid = (i < n_var);
    float f0 = 0.0f, f1 = 0.0f;
    int isCon = 0;
    if (valid) {
      long long node = assoc_var[i];
      int ov = owner[node];
      long long src = (long long)(ov & 0x3FFFFFFF);
      isCon = (ov >> 30) & 1;
      const float* F = isCon ? conF : varF;
      f0 = F[2 * src];
      f1 = F[2 * src + 1];
    }
    // ---- layer 0 (K=2, VALU, float4): h1 = relu(f @ Wa1 + ba1) -------------
    const float* Wa = isCon ? cW1 : vW1;
    const float* ba = isCon ? cb1 : vb1;
#pragma unroll
    for (int j = 0; j < 16; ++j) {
      int c = half * 64 + j * 4;
      float4 w0 = *(const float4*)&Wa[c];
      float4 w1 = *(const float4*)&Wa[DIM + c];
      float4 bb = *(const float4*)&ba[c];
      float4 h = {0.0f, 0.0f, 0.0f, 0.0f};
      if (valid) {
        h.x = fmaxf(f0 * w0.x + f1 * w1.x + bb.x, 0.0f);
        h.y = fmaxf(f0 * w0.y + f1 * w1.y + bb.y, 0.0f);
        h.z = fmaxf(f0 * w0.z + f1 * w1.z + bb.z, 0.0f);
        h.w = fmaxf(f0 * w0.w + f1 * w1.w + bb.w, 0.0f);
      }
      *(float4*)&buf[n * LSTR + c] = h;
    }

    // ---- wave-uniform tile type (EXEC stays full at every WMMA) ------------
    unsigned conMask = (unsigned)(__ballot(isCon) & 0xFFFFull);
    bool hasCon = (conMask != 0u);
    bool hasVar = __any((int)(valid && !isCon)) != 0;

    // ---- embedding layer 2 (no relu) ---------------------------------------
    if (hasCon && hasVar)
      layer_wmma_mixed(buf, buf, Pv2, vb2, Pc2, cb2, n, half, lane, conMask);
    else if (hasCon)
      layer_wmma(buf, buf, Pc2, cb2, n, half, lane, false);
    else
      layer_wmma(buf, buf, Pv2, vb2, n, half, lane, false);

    // ---- head --------------------------------------------------------------
    layer_wmma(buf, buf, P1, b1, n, half, lane, true);
    layer_wmma(buf, buf, P2, b2, n, half, lane, true);
    layer_wmma(buf, buf, P3, b3, n, half, lane, true);

    // ---- final 128->1 dot + sigmoid (split across half-waves) --------------
    float s = 0.0f;
#pragma unroll
    for (int j = 0; j < 16; ++j) {
      int c = half * 64 + j * 4;
      float4 x = *(const float4*)&buf[n * LSTR + c];
      float4 w = *(const float4*)&W4[c];
      s += x.x * w.x + x.y * w.y + x.z * w.z + x.w * w.w;
    }
    s += __shfl_xor(s, 16);
    if (half == 0 && valid)
      out[i] = 1.0f / (1.0f + expf(-(s + b4[0])));
  }
}

extern "C" void kernel_launch(void* const* d_in, const int* in_sizes, int n_in,
                              void* d_out, int out_size, void* d_ws, size_t ws_size,
                              hipStream_t stream) {
  const float* varF = (const float*)d_in[0];
  const float* conF = (const float*)d_in[1];
  // d_in[2] = node_types (unused by the reference computation)
  const long long* assoc_var = (const long long*)d_in[3];
  const long long* assoc_con = (const long long*)d_in[4];
  const float* vW1 = (const float*)d_in[5];
  const float* vb1 = (const float*)d_in[6];
  const float* vW2 = (const float*)d_in[7];
  const float* vb2 = (const float*)d_in[8];
  const float* cW1 = (const float*)d_in[9];
  const float* cb1 = (const float*)d_in[10];
  const float* cW2 = (const float*)d_in[11];
  const float* cb2 = (const float*)d_in[12];
  const float* W1  = (const float*)d_in[13];
  const float* b1  = (const float*)d_in[14];
  const float* W2  = (const float*)d_in[15];
  const float* b2  = (const float*)d_in[16];
  const float* W3  = (const float*)d_in[17];
  const float* b3  = (const float*)d_in[18];
  const float* W4  = (const float*)d_in[19];
  const float* b4  = (const float*)d_in[20];
  float* out = (float*)d_out;

  const int n_var   = in_sizes[0] / 2;
  const int n_con   = in_sizes[1] / 2;
  const int n_nodes = in_sizes[2];

  // workspace: [ owner: n_nodes int32 | 5 packed weight matrices ]
  int* owner = (int*)d_ws;
  size_t ownerBytes = (((size_t)n_nodes * 4) + 255) & ~(size_t)255;
  float* packed = (float*)((char*)d_ws + ownerBytes);
  float* Pv2 = packed + 0 * WMAT;
  float* Pc2 = packed + 1 * WMAT;
  float* P1  = packed + 2 * WMAT;
  float* P2  = packed + 3 * WMAT;
  float* P3  = packed + 4 * WMAT;

  owner_scatter_var<<<(n_var + 255) / 256, 256, 0, stream>>>(assoc_var, owner, n_var);
  owner_scatter_con<<<(n_con + 255) / 256, 256, 0, stream>>>(assoc_con, owner, n_con);
  pack_w<<<WMAT / 256, 256, 0, stream>>>(vW2, Pv2);
  pack_w<<<WMAT / 256, 256, 0, stream>>>(cW2, Pc2);
  pack_w<<<WMAT / 256, 256, 0, stream>>>(W1,  P1);
  pack_w<<<WMAT / 256, 256, 0, stream>>>(W2,  P2);
  pack_w<<<WMAT / 256, 256, 0, stream>>>(W3,  P3);

  const int n_tiles = (n_var + 15) / 16;
  const int blocks = (n_tiles + WPB - 1) / WPB;
  fused_net<<<blocks, WPB * 32, 0, stream>>>(
      varF, conF, assoc_var, owner,
      vW1, vb1, Pv2, vb2, cW1, cb1, Pc2, cb2,
      P1, b1, P2, b2, P3, b3, W4, b4,
      out, n_var, n_tiles);
}